// BANLayer_89489938579682
// MI455X (gfx1250) — compile-verified
//
#include <hip/hip_runtime.h>
#include <hip/hip_bf16.h>
#include <math.h>

// ---------------------------------------------------------------------------
// BANLayer forward for MI455X (gfx1250, wave32).
// fp32 matrix pipe (V_WMMA_F32_16X16X4_F32) for all GEMM stages.
// Global->LDS staging is done by the Tensor Data Mover (tensor_load_to_lds,
// TENSORcnt / s_wait_tensorcnt), double-buffered against the WMMA work.
// TDM hardware padding (32 DW interval, +4 DW) recreates the 36-float LDS row
// stride -> 16B-aligned rows + bank-conflict-free ds_load_b64 fragments.
// ---------------------------------------------------------------------------

typedef __attribute__((ext_vector_type(2))) float    v2f;
typedef __attribute__((ext_vector_type(8))) float    v8f;
typedef __attribute__((ext_vector_type(4))) unsigned v4u;
typedef __attribute__((ext_vector_type(4))) int      v4i;
typedef __attribute__((ext_vector_type(8))) int      v8i;

#define WMMA_F32(a, b, c) \
  __builtin_amdgcn_wmma_f32_16x16x4_f32(false, (a), false, (b), (short)0, (c), false, false)

namespace {
constexpr int B    = 16;
constexpr int VN   = 256;
constexpr int QN   = 256;
constexpr int VD   = 512;
constexpr int QD   = 768;
constexpr int HOUT = 8;
constexpr int HD   = 512;
constexpr int KP   = 3;
constexpr int HK   = HD * KP;  // 1536

constexpr int KC  = 32;        // k-chunk per TDM transfer
constexpr int LS  = 36;        // LDS row stride in floats (32 data + 4 pad)
constexpr int ASZ = 128 * LS;  // one A buffer (floats)
constexpr int BSZ = 64 * LS;   // one B buffer (floats)
}

// ----------------------------------------------------------------------------
// TDM: load a [rows x width] fp32 tile (row stride `stride` elements) from
// global into LDS at byte offset lds_off, inserting 4 DWORDs of padding after
// every 32 DWORDs stored (-> LDS row stride = 36 floats).
// D# layout per CDNA5 ISA ch.8 (normal mode, 2D, no gather/iterate).
// ----------------------------------------------------------------------------
__device__ __forceinline__ void tdm_load_tile_f32(unsigned lds_off,
                                                  const float* gptr, int width,
                                                  int rows, int stride) {
  const unsigned long long ga = (unsigned long long)gptr;
  v4u g0;
  g0[0] = 1u;                                   // count=1 (valid), user mode
  g0[1] = lds_off;                              // lds_addr (bytes)
  g0[2] = (unsigned)(ga & 0xFFFFFFFFu);         // global_addr[31:0]
  g0[3] = (unsigned)((ga >> 32) & 0x01FFFFFFu)  // global_addr[56:32]
          | (2u << 30);                         // type = 2 ("image")
  v8i g1;
  g1[0] = (2 << 16)     // data_size = 4B
          | (1 << 20)   // pad_enable
          | (4 << 22)   // pad_interval code 4 -> every 32 DWORDs
          | (3 << 25);  // pad_amount  code 3 -> +4 DWORDs
  g1[1] = (width & 0xFFFF) << 16;                            // tensor_dim0 lo
  g1[2] = ((width >> 16) & 0xFFFF) | ((rows & 0xFFFF) << 16);  // d0 hi | d1 lo
  g1[3] = ((rows >> 16) & 0xFFFF) | ((width & 0xFFFF) << 16);  // d1 hi | tile_dim0
  g1[4] = rows & 0xFFFF;  // tile_dim1 (tile_dim2 = 0)
  g1[5] = stride;         // tensor_dim0_stride[31:0]
  g1[6] = 0;              // stride[47:32] | tensor_dim1_stride lo (unused, 2D)
  g1[7] = 0;
  const v4i gz = {0, 0, 0, 0};
#if defined(__clang_major__) && __clang_major__ >= 23
  const v8i gz8 = {0, 0, 0, 0, 0, 0, 0, 0};
  __builtin_amdgcn_tensor_load_to_lds(g0, g1, gz, gz, gz8, 0);
#else
  __builtin_amdgcn_tensor_load_to_lds(g0, g1, gz, gz, 0);
#endif
}

// ----------------------------------------------------------------------------
// Block core: acc[4] (16x64 per wave, 128x64 per block) += X[128,D] @ W[64,D]^T
// TDM double-buffers 32-wide k-chunks; wave 0 drives the DMA; fragments are
// register-double-buffered across the 8 k-steps of each chunk.
// SCALED: multiply A fragments by scale[k] in-register (att head weights).
// ----------------------------------------------------------------------------
template <bool SCALED>
__device__ __forceinline__ void gemm_core_128x64(
    const float* __restrict__ Xblk, const float* __restrict__ Wblk,
    const float* __restrict__ scale, int D,
    float* __restrict__ ldsA, float* __restrict__ ldsB, v8f acc[4]) {
  const int t = threadIdx.x;
  const int lane = t & 31;
  const int warp = t >> 5;
  const int lo = lane & 15;
  const int hi = lane >> 4;
  const int kb = hi * 2;  // f32 A/B frag: lanes 0-15 -> K=0,1; 16-31 -> K=2,3
  const int NC = D / KC;

  const unsigned offA = (unsigned)(unsigned long long)ldsA;
  const unsigned offB = (unsigned)(unsigned long long)ldsB;

  if (warp == 0) {  // prologue DMA: chunk 0 -> buffer 0
    tdm_load_tile_f32(offA, Xblk, KC, 128, D);
    tdm_load_tile_f32(offB, Wblk, KC, 64, D);
  }

  for (int c = 0; c < NC; ++c) {
    if (warp == 0) __builtin_amdgcn_s_wait_tensorcnt(0);  // chunk c landed
    __syncthreads();  // LDS visible; all waves done reading buffer (c+1)&1
    if (warp == 0 && c + 1 < NC) {  // DMA chunk c+1 while we crunch chunk c
      const int k0 = (c + 1) * KC;
      const unsigned bsel = (unsigned)((c + 1) & 1);
      tdm_load_tile_f32(offA + bsel * (ASZ * 4), Xblk + k0, KC, 128, D);
      tdm_load_tile_f32(offB + bsel * (BSZ * 4), Wblk + k0, KC, 64, D);
    }

    // ---- 8 k-steps x 4 WMMA from LDS buffer c&1, fragments pipelined ----
    const float* Ar = ldsA + (size_t)(c & 1) * ASZ + (warp * 16 + lo) * LS + kb;
    const float* Br = ldsB + (size_t)(c & 1) * BSZ + lo * LS + kb;
    const float* Hr = scale + c * KC + kb;  // only read when SCALED
    v2f aC  = *(const v2f*)(Ar);
    v2f b0C = *(const v2f*)(Br);
    v2f b1C = *(const v2f*)(Br + 16 * LS);
    v2f b2C = *(const v2f*)(Br + 32 * LS);
    v2f b3C = *(const v2f*)(Br + 48 * LS);
    v2f hC = {};
    if (SCALED) hC = *(const v2f*)(Hr);
#pragma unroll
    for (int ks = 0; ks < KC / 4; ++ks) {
      v2f aN = aC, b0N = b0C, b1N = b1C, b2N = b2C, b3N = b3C, hN = hC;
      if (ks < KC / 4 - 1) {
        const int cc = (ks + 1) * 4;
        aN  = *(const v2f*)(Ar + cc);
        b0N = *(const v2f*)(Br + cc);
        b1N = *(const v2f*)(Br + 16 * LS + cc);
        b2N = *(const v2f*)(Br + 32 * LS + cc);
        b3N = *(const v2f*)(Br + 48 * LS + cc);
        if (SCALED) hN = *(const v2f*)(Hr + cc);
      }
      v2f aS = aC;
      if (SCALED) {
        aS[0] = aC[0] * hC[0];
        aS[1] = aC[1] * hC[1];
      }
      acc[0] = WMMA_F32(aS, b0C, acc[0]);
      acc[1] = WMMA_F32(aS, b1C, acc[1]);
      acc[2] = WMMA_F32(aS, b2C, acc[2]);
      acc[3] = WMMA_F32(aS, b3C, acc[3]);
      aC = aN; b0C = b0N; b1C = b1N; b2C = b2N; b3C = b3N; hC = hN;
    }
  }
}

// ----------------------------------------------------------------------------
// Y[M,N] = relu(X[M,D] @ W[N,D]^T + bias[N]);  block tile 128M x 64N
// ----------------------------------------------------------------------------
__global__ __launch_bounds__(256) void fc_relu_kernel(
    const float* __restrict__ X, const float* __restrict__ W,
    const float* __restrict__ bias, float* __restrict__ Y, int N, int D) {
  __shared__ float ldsA[2 * ASZ];
  __shared__ float ldsB[2 * BSZ];
  const int lane = threadIdx.x & 31;
  const int warp = threadIdx.x >> 5;
  const int lo = lane & 15;
  const int hi = lane >> 4;
  const int m0 = blockIdx.y * 128;
  const int n0 = blockIdx.x * 64;

  v8f acc[4] = {};
  gemm_core_128x64<false>(X + (size_t)m0 * D, W + (size_t)n0 * D, nullptr, D,
                          ldsA, ldsB, acc);

  const int mrow0 = m0 + warp * 16;
#pragma unroll
  for (int tt = 0; tt < 4; ++tt) {
    const int ncol = n0 + tt * 16 + lo;
    const float bn = bias[ncol];
#pragma unroll
    for (int j = 0; j < 8; ++j) {  // D layout: row = j + 8*hi, col = lo
      const float c = acc[tt][j] + bn;
      Y[(size_t)(mrow0 + j + 8 * hi) * N + ncol] = c > 0.f ? c : 0.f;
    }
  }
}

// ----------------------------------------------------------------------------
// att[b,h,v,q] = sum_k (v_[b,v,k]*h_mat[h,k]) * q_[b,q,k] + h_bias[h], masked.
// ----------------------------------------------------------------------------
__global__ __launch_bounds__(256) void att_kernel(
    const float* __restrict__ Vh, const float* __restrict__ Qh,
    const float* __restrict__ h_mat, const float* __restrict__ h_bias,
    const int* __restrict__ vz, const int* __restrict__ qz,
    float* __restrict__ att) {
  __shared__ float ldsA[2 * ASZ];
  __shared__ float ldsB[2 * BSZ];
  const int lane = threadIdx.x & 31;
  const int warp = threadIdx.x >> 5;
  const int lo = lane & 15;
  const int hi = lane >> 4;
  const int bh = blockIdx.z;  // b*HOUT + h
  const int b = bh >> 3;
  const int h = bh & 7;
  const int v0 = blockIdx.y * 128;
  const int q0 = blockIdx.x * 64;

  v8f acc[4] = {};
  gemm_core_128x64<true>(Vh + ((size_t)b * VN + v0) * HK,
                         Qh + ((size_t)b * QN + q0) * HK,
                         h_mat + (size_t)h * HK, HK, ldsA, ldsB, acc);

  const float hb = h_bias[h];
  const int vrow0 = v0 + warp * 16;
#pragma unroll
  for (int tt = 0; tt < 4; ++tt) {
    const int qcol = q0 + tt * 16 + lo;
    const int qm = qz[b * QN + qcol];
#pragma unroll
    for (int j = 0; j < 8; ++j) {
      const int vrow = vrow0 + j + 8 * hi;
      float val = acc[tt][j] + hb;
      if (vz[b * VN + vrow] || qm) val = -INFINITY;
      att[((size_t)bh * VN + vrow) * QN + qcol] = val;
    }
  }
}

// ----------------------------------------------------------------------------
// zf[r] = (sum_d |X[r,d]| == 0)
// ----------------------------------------------------------------------------
__global__ void rowzero_kernel(const float* __restrict__ X, int rows, int D,
                               int* __restrict__ zf) {
  const int r = blockIdx.x * blockDim.x + threadIdx.x;
  if (r >= rows) return;
  const float* p = X + (size_t)r * D;
  float s = 0.f;
  for (int d = 0; d < D; ++d) s += fabsf(p[d]);
  zf[r] = (s == 0.f) ? 1 : 0;
}

// ----------------------------------------------------------------------------
// A_sum[b,v,q] = sum_h att[b,h,v,q]
// ----------------------------------------------------------------------------
__global__ void att_sumh_kernel(const float* __restrict__ att,
                                float* __restrict__ As) {
  const size_t idx = (size_t)blockIdx.x * blockDim.x + threadIdx.x;
  const int b = (int)(idx >> 16);
  const int vq = (int)(idx & 0xFFFF);
  const float* p = att + (size_t)b * HOUT * VN * QN + vq;
  float s = 0.f;
#pragma unroll
  for (int h = 0; h < HOUT; ++h) s += p[(size_t)h * VN * QN];
  As[idx] = s;
}

// ----------------------------------------------------------------------------
// logits_k[b,k] = sum_v v_[b,v,k] * (A_sum[b] @ q_[b])[v,k]
// ----------------------------------------------------------------------------
__global__ __launch_bounds__(256) void pool_kernel(
    const float* __restrict__ As, const float* __restrict__ Qh,
    const float* __restrict__ Vh, float* __restrict__ logits_k) {
  const int lane = threadIdx.x & 31;
  const int warp = threadIdx.x >> 5;
  const int lo = lane & 15;
  const int hi = lane >> 4;
  const int kb = hi * 2;

  const int b = blockIdx.y;
  const int ktile = blockIdx.x * 8 + warp;  // 96 k-tiles
  const int kcol = ktile * 16 + lo;

  float total = 0.f;
  for (int vt = 0; vt < VN / 16; ++vt) {
    v8f acc = {};
    const float* Arow = As + ((size_t)b * VN + vt * 16 + lo) * QN + kb;
    for (int qk = 0; qk < QN; qk += 4) {
      v2f a = *(const v2f*)(Arow + qk);
      const float* Bp = Qh + ((size_t)b * QN + qk + kb) * HK + kcol;
      v2f bb;
      bb[0] = Bp[0];
      bb[1] = Bp[HK];
      acc = WMMA_F32(a, bb, acc);
    }
    const float* vp = Vh + ((size_t)b * VN + vt * 16 + 8 * hi) * HK + kcol;
#pragma unroll
    for (int j = 0; j < 8; ++j) total += acc[j] * vp[(size_t)j * HK];
  }
  total += __shfl_xor(total, 16, 32);  // fold the two row-halves (same kcol)
  if (hi == 0) logits_k[(size_t)b * HK + kcol] = total;
}

// ----------------------------------------------------------------------------
// k-pool (sum groups of 3) + BatchNorm1d with batch statistics.
// ----------------------------------------------------------------------------
__global__ void bn_kernel(const float* __restrict__ logits_k,
                          const float* __restrict__ gamma,
                          const float* __restrict__ beta,
                          float* __restrict__ out) {
  const int j = blockIdx.x * blockDim.x + threadIdx.x;
  if (j >= HD) return;
  float x[B];
  float mu = 0.f;
#pragma unroll
  for (int b = 0; b < B; ++b) {
    const float* p = logits_k + (size_t)b * HK + j * KP;
    const float s = p[0] + p[1] + p[2];
    x[b] = s;
    mu += s;
  }
  mu *= (1.f / B);
  float var = 0.f;
#pragma unroll
  for (int b = 0; b < B; ++b) {
    const float d = x[b] - mu;
    var += d * d;
  }
  var *= (1.f / B);
  const float inv = rsqrtf(var + 1e-5f) * gamma[j];
  const float bt = beta[j];
#pragma unroll
  for (int b = 0; b < B; ++b) out[(size_t)b * HD + j] = (x[b] - mu) * inv + bt;
}

// ----------------------------------------------------------------------------
extern "C" void kernel_launch(void* const* d_in, const int* in_sizes, int n_in,
                              void* d_out, int out_size, void* d_ws,
                              size_t ws_size, hipStream_t stream) {
  const float* v      = (const float*)d_in[0];  // [16,256,512]
  const float* q      = (const float*)d_in[1];  // [16,256,768]
  const float* Wv     = (const float*)d_in[2];  // [1536,512]
  const float* bv     = (const float*)d_in[3];  // [1536]
  const float* Wq     = (const float*)d_in[4];  // [1536,768]
  const float* bq     = (const float*)d_in[5];  // [1536]
  const float* h_mat  = (const float*)d_in[6];  // [8,1536]
  const float* h_bias = (const float*)d_in[7];  // [8]
  const float* gamma  = (const float*)d_in[8];  // [512]
  const float* beta   = (const float*)d_in[9];  // [512]

  float* out        = (float*)d_out;
  float* out_logits = out;         // [16,512]
  float* out_att    = out + 8192;  // [16,8,256,256]

  // Workspace (floats): v_ | q_ | A_sum | logits_k | vz | qz  (~54.7MB)
  float* ws       = (float*)d_ws;
  float* Vh       = ws;                                 // 16*256*1536
  float* Qh       = Vh + (size_t)B * VN * HK;           // 16*256*1536
  float* As       = Qh + (size_t)B * QN * HK;           // 16*256*256
  float* logits_k = As + (size_t)B * VN * QN;           // 16*1536
  int*   vz       = (int*)(logits_k + (size_t)B * HK);  // 16*256
  int*   qz       = vz + B * VN;                        // 16*256

  // 1) v_ = relu(v @ Wv^T + bv):  M=4096, N=1536, D=512
  fc_relu_kernel<<<dim3(HK / 64, (B * VN) / 128), 256, 0, stream>>>(
      v, Wv, bv, Vh, HK, VD);
  // 2) q_ = relu(q @ Wq^T + bq):  M=4096, N=1536, D=768
  fc_relu_kernel<<<dim3(HK / 64, (B * QN) / 128), 256, 0, stream>>>(
      q, Wq, bq, Qh, HK, QD);

  // 3) zero-row masks
  rowzero_kernel<<<(B * VN) / 256, 256, 0, stream>>>(v, B * VN, VD, vz);
  rowzero_kernel<<<(B * QN) / 256, 256, 0, stream>>>(q, B * QN, QD, qz);

  // 4) att (written directly to output, masked, bias added)
  att_kernel<<<dim3(QN / 64, VN / 128, B * HOUT), 256, 0, stream>>>(
      Vh, Qh, h_mat, h_bias, vz, qz, out_att);

  // 5) A_sum = sum over heads of att
  att_sumh_kernel<<<(B * VN * QN) / 256, 256, 0, stream>>>(out_att, As);

  // 6) logits_k[b,k] = sum_v v_ * (A_sum @ q_)
  pool_kernel<<<dim3((HK / 16) / 8, B), 256, 0, stream>>>(As, Qh, Vh, logits_k);

  // 7) k-pool + batchnorm
  bn_kernel<<<(HD + 255) / 256, 256, 0, stream>>>(logits_k, gamma, beta,
                                                  out_logits);
}